// RelationalGraphConvLayer_57947698758346
// MI455X (gfx1250) — compile-verified
//
#include <hip/hip_runtime.h>

typedef float v2f __attribute__((ext_vector_type(2)));
typedef float v8f __attribute__((ext_vector_type(8)));

#define AS1 __attribute__((address_space(1)))
#define AS3 __attribute__((address_space(3)))

namespace {
constexpr int kB     = 256;
constexpr int kBond  = 5;
constexpr int kN     = 128;
constexpr int kAtom  = 64;
constexpr int kUnits = 128;

// All strides ≡ 4 (mod 64): the WMMA b64 operand fetch (lanes 0-15 read
// cols kk,kk+1; lanes 16-31 read kk+2,kk+3 of rows ln*stride) then covers
// banks {4*ln+kk .. +1} ∪ {4*ln+kk+2 .. +1} = a perfect 64-bank permutation.
constexpr int kFtStride = 132;  // feat^T : [64][132]  (d-major rows, m contiguous)
constexpr int kXStride  = 68;   // X      : [128][68]  (m rows, d contiguous)
constexpr int kKtStride = 68;   // K^T    : [128][68]  (u-major rows, d contiguous)
}

__launch_bounds__(256)
__global__ void rgcn_fused_wmma(const float* __restrict__ adj,
                                const float* __restrict__ feat,
                                const float* __restrict__ ker,
                                float* __restrict__ out)
{
    __shared__ float feat_t[kAtom * kFtStride];  // 33.0 KB
    __shared__ float x_s[kN * kXStride];         // 34.0 KB
    __shared__ float k_t[kUnits * kKtStride];    // 34.0 KB

    const int b    = blockIdx.x;
    const int tid  = threadIdx.x;
    const int wave = tid >> 5;        // 0..7
    const int lane = tid & 31;
    const int ln   = lane & 15;
    const int half = lane >> 4;       // k-offset select (0 or 2)

    // ---- features -> LDS, transposed (coalesced global read) ----
    {
        const float* src = feat + (size_t)b * (kN * kAtom);
        #pragma unroll
        for (int i = 0; i < (kN * kAtom) / 256; ++i) {
            const int idx = tid + i * 256;
            const int m = idx >> 6, d = idx & 63;
            feat_t[d * kFtStride + m] = src[idx];
        }
    }

    // persistent output accumulators: 8 tiles of 16x16 fp32 per wave
    v8f acc[8];
    #pragma unroll
    for (int t = 0; t < 8; ++t)
        #pragma unroll
        for (int r = 0; r < 8; ++r)
            acc[t][r] = 0.0f;

    const int mrow = wave * 16 + ln;   // A-operand row owned by this lane

    for (int e = 0; e < kBond; ++e) {
        __syncthreads();  // prior iteration's readers of k_t / x_s are done

        // ---- K_e -> LDS transposed; async so it overlaps stage-1 compute ----
        {
            const float* src = ker + (size_t)(b * kBond + e) * (kAtom * kUnits);
            #pragma unroll
            for (int i = 0; i < (kAtom * kUnits) / 256; ++i) {
                const int idx = tid + i * 256;
                const int d = idx >> 7, u = idx & 127;
#if __has_builtin(__builtin_amdgcn_global_load_async_to_lds_b32)
                __builtin_amdgcn_global_load_async_to_lds_b32(
                    (AS1 int*)(AS1 void*)(src + idx),
                    (AS3 int*)(AS3 void*)&k_t[u * kKtStride + d],
                    0, 0);
#else
                k_t[u * kKtStride + d] = src[idx];
#endif
            }
        }

        // ---- stage 1: X[128x64] = adj_e[128x128] @ feat[128x64] ----
        {
            const float* arow =
                adj + ((size_t)(b * kBond + e) * kN + mrow) * kN;

            v8f xacc[4];
            #pragma unroll
            for (int t = 0; t < 4; ++t)
                #pragma unroll
                for (int r = 0; r < 8; ++r)
                    xacc[t][r] = 0.0f;

            #pragma unroll 4
            for (int mk = 0; mk < kN; mk += 4) {
                const int kk = mk + 2 * half;
                v2f a = *(const v2f*)(arow + kk);              // A[m][kk..kk+1]
                #pragma unroll
                for (int td = 0; td < 4; ++td) {
                    v2f bm = *(const v2f*)(
                        &feat_t[(td * 16 + ln) * kFtStride + kk]);  // B[kk..kk+1][n]
                    xacc[td] = __builtin_amdgcn_wmma_f32_16x16x4_f32(
                        false, a, false, bm, (short)0, xacc[td], false, false);
                }
            }

            // C-layout -> LDS (transpose into A-operand layout for stage 2)
            #pragma unroll
            for (int td = 0; td < 4; ++td)
                #pragma unroll
                for (int r = 0; r < 8; ++r)
                    x_s[(wave * 16 + half * 8 + r) * kXStride + td * 16 + ln] =
                        xacc[td][r];
        }

#if __has_builtin(__builtin_amdgcn_global_load_async_to_lds_b32)
        asm volatile("s_wait_asynccnt 0" ::: "memory");
#endif
        __syncthreads();  // x_s stores + K_e async copy complete, all waves

        // ---- stage 2: acc += X[128x64] @ K_e[64x128] ----
        #pragma unroll 4
        for (int mk = 0; mk < kAtom; mk += 4) {
            const int kk = mk + 2 * half;
            v2f a = *(const v2f*)(&x_s[mrow * kXStride + kk]);  // A[m][kk..kk+1]
            #pragma unroll
            for (int tu = 0; tu < 8; ++tu) {
                v2f bm = *(const v2f*)(
                    &k_t[(tu * 16 + ln) * kKtStride + kk]);     // B[kk..kk+1][u]
                acc[tu] = __builtin_amdgcn_wmma_f32_16x16x4_f32(
                    false, a, false, bm, (short)0, acc[tu], false, false);
            }
        }
    }

    // ---- ReLU + store ----
    float* ob = out + (size_t)b * (kN * kUnits);
    #pragma unroll
    for (int tu = 0; tu < 8; ++tu)
        #pragma unroll
        for (int r = 0; r < 8; ++r) {
            const int m = wave * 16 + half * 8 + r;
            ob[m * kUnits + tu * 16 + ln] = fmaxf(acc[tu][r], 0.0f);
        }
}

extern "C" void kernel_launch(void* const* d_in, const int* in_sizes, int n_in,
                              void* d_out, int out_size, void* d_ws, size_t ws_size,
                              hipStream_t stream) {
    const float* adj  = (const float*)d_in[0];  // (256, 5, 128, 128)
    const float* feat = (const float*)d_in[1];  // (256, 128, 64)
    const float* ker  = (const float*)d_in[2];  // (256, 5, 64, 128)
    float* out        = (float*)d_out;          // (256, 128, 128)

    dim3 grid(kB);     // one workgroup per batch element
    dim3 block(256);   // 8 wave32 waves
    hipLaunchKernelGGL(rgcn_fused_wmma, grid, block, 0, stream,
                       adj, feat, ker, out);
}